// TTN_Conv_2by2to1_40638980555114
// MI455X (gfx1250) — compile-verified
//
#include <hip/hip_runtime.h>
#include <hip/hip_bf16.h>

typedef __attribute__((ext_vector_type(2))) float v2f;
typedef __attribute__((ext_vector_type(8))) float v8f;

#define NUM   128
#define CIN   3
#define COUT  8
#define NX    31
#define NY    31
#define DIN   4
#define DOUT  6
#define SX    32
#define SY    32
// per-c weight chunk: COUT * 256 * DOUT floats = 12288 floats = 48 KB
#define WCHUNK_F   (COUT * 256 * DOUT)
#define WCHUNK_F4  (WCHUNK_F / 4)        // 3072 float4
#define PSTRIDE_F  (256 * DOUT)          // 1536 floats per (c,p) chunk
#define PSTRIDE_F4 (PSTRIDE_F / 4)       // 384

__launch_bounds__(256, 1)
__global__ void ttn_conv_2x2_wmma_kernel(const float* __restrict__ x,
                                         const float* __restrict__ tensors,
                                         const float* __restrict__ bias,
                                         float* __restrict__ out) {
    extern __shared__ float smem[];      // 48 KB: weight chunk for current c

    const int px   = blockIdx.x;         // 0..30
    const int py   = blockIdx.y;         // 0..30
    const int tid  = threadIdx.x;
    const int lane = tid & 31;
    const int wv   = tid >> 5;           // 0..7 -> M tile (16 batch rows)
    const int lrow = lane & 15;
    const bool hi  = (lane >= 16);       // f32-WMMA A layout: low half K=0,1; high half K=2,3

    const int n      = wv * 16 + lrow;   // batch row this lane feeds
    const int xybase = px * SY + py;

    // ---- Build per-lane rank-1 factors of the patch row ----
    // patch(n, c*256 + m) with m = ((i*4+j)*4+k)*4+l  ->  ab[c][i*4+j] * (xc[k]*xd[l])
    float ab[CIN][16];
    float cdp0[CIN][4];                  // xc[q] * xd[off]
    float cdp1[CIN][4];                  // xc[q] * xd[off+1]
    #pragma unroll
    for (int c = 0; c < CIN; ++c) {
        float xa[4], xb[4], xc[4], xd[4];
        #pragma unroll
        for (int i = 0; i < 4; ++i) {
            const float* base = x + (size_t)((n * CIN + c) * DIN + i) * (SX * SY) + xybase;
            xa[i] = base[0];             // (x  , y  )
            xb[i] = base[SY];            // (x+1, y  )
            xc[i] = base[1];             // (x  , y+1)
            xd[i] = base[SY + 1];        // (x+1, y+1)
        }
        #pragma unroll
        for (int i = 0; i < 4; ++i)
            #pragma unroll
            for (int j = 0; j < 4; ++j)
                ab[c][i * 4 + j] = xa[i] * xb[j];
        const float d0 = hi ? xd[2] : xd[0];
        const float d1 = hi ? xd[3] : xd[1];
        #pragma unroll
        for (int q = 0; q < 4; ++q) {
            cdp0[c][q] = xc[q] * d0;
            cdp1[c][q] = xc[q] * d1;
        }
    }

    // ---- B-operand LDS base per N-tile (fold lane-half K offset into address) ----
    int bb[3];
    #pragma unroll
    for (int nt = 0; nt < 3; ++nt) {
        const int nn = nt * 16 + lrow;       // 0..47
        const int p  = nn / 6;
        const int o  = nn - p * 6;
        bb[nt] = p * PSTRIDE_F + o + (hi ? 2 : 0) * DOUT;
    }

    v8f acc[3] = {};

    const float4* gt4 = reinterpret_cast<const float4*>(tensors);
    float4* sm4 = reinterpret_cast<float4*>(smem);

    #pragma unroll
    for (int c = 0; c < CIN; ++c) {
        __syncthreads();
        // ---- Stage 48 KB weight chunk for this c: lds[p*1536 + m*6+o] ----
        #pragma unroll
        for (int r = 0; r < 12; ++r) {
            const int flat4 = r * 256 + tid;             // 0..3071
            const int p     = flat4 / PSTRIDE_F4;        // 0..7
            const int q     = flat4 - p * PSTRIDE_F4;    // 0..383
            const size_t g4 = (size_t)(((c * COUT + p) * NX + px) * NY + py) * PSTRIDE_F4 + q;
            sm4[flat4] = gt4[g4];
        }
        __syncthreads();

        // ---- 64 K-steps of 16x16x4 f32 WMMA, 3 N-tiles each ----
        #pragma unroll
        for (int m4 = 0; m4 < 64; ++m4) {
            const float abv = ab[c][m4 >> 2];
            v2f A;
            A.x = abv * cdp0[c][m4 & 3];
            A.y = abv * cdp1[c][m4 & 3];
            #pragma unroll
            for (int nt = 0; nt < 3; ++nt) {
                v2f B;
                B.x = smem[bb[nt] + m4 * 24];
                B.y = smem[bb[nt] + m4 * 24 + 6];
                acc[nt] = __builtin_amdgcn_wmma_f32_16x16x4_f32(
                    /*neg_a=*/false, A, /*neg_b=*/false, B,
                    /*c_mod=*/(short)0, acc[nt],
                    /*reuse_a=*/false, /*reuse_b=*/false);
            }
        }
    }

    // ---- Epilogue: add bias, scatter to out[n, p, o, x, y] ----
    const int obase = px * NY + py;
    #pragma unroll
    for (int nt = 0; nt < 3; ++nt) {
        const int nn = nt * 16 + lrow;
        const int p  = nn / 6;
        const int o  = nn - p * 6;
        const float bv = bias[((p * NX + px) * NY + py) * DOUT + o];
        #pragma unroll
        for (int r = 0; r < 8; ++r) {
            const int row = wv * 16 + r + (hi ? 8 : 0);   // C/D layout: VGPR r -> M=r (lo) / M=r+8 (hi)
            out[(size_t)((row * COUT + p) * DOUT + o) * (NX * NY) + obase] = acc[nt][r] + bv;
        }
    }
}

extern "C" void kernel_launch(void* const* d_in, const int* in_sizes, int n_in,
                              void* d_out, int out_size, void* d_ws, size_t ws_size,
                              hipStream_t stream) {
    const float* x       = (const float*)d_in[0];
    const float* tensors = (const float*)d_in[1];
    const float* bias    = (const float*)d_in[2];
    float* out           = (float*)d_out;

    dim3 grid(NX, NY, 1);
    dim3 block(256, 1, 1);
    const size_t lds_bytes = WCHUNK_F * sizeof(float);   // 48 KB dynamic LDS
    ttn_conv_2x2_wmma_kernel<<<grid, block, lds_bytes, stream>>>(x, tensors, bias, out);
}